// SNN_34565896798641
// MI455X (gfx1250) — compile-verified
//
#include <hip/hip_runtime.h>

// ---------------------------------------------------------------------------
// SNN forward, fully fused for MI455X (gfx1250, wave32, WMMA).
//   B=8192, D=H=256, OUT=1, T=64.
//   Kernel 1: cur_in = x[:, :, 0] @ W_in^T + b_in        (f32 WMMA, W_in in LDS)
//   Kernel 2: per 32-row batch tile, all 64 timesteps fused:
//             spikes -> WMMA GEMM vs LDS-resident W_h -> m2/m3 recurrences
//             -> batch-mean via atomicAdd.
// Weights live in LDS as K-pair-interleaved float2 so every WMMA B fragment
// is one conflict-free ds_load_b64 into an aligned VGPR pair.
// ---------------------------------------------------------------------------

typedef __attribute__((ext_vector_type(2))) float v2f;
typedef __attribute__((ext_vector_type(8))) float v8f;

#define B_TOT      8192
#define DIM        256
#define HID        256
#define TSTEPS     64
#define TILE_ROWS  32
#define WT2_STRIDE 544   // floats per k2 row (mod 64 == 32 -> halves hit disjoint banks)
#define BUF_STRIDE 260   // padded row stride for the 32x256 activation tile

__device__ __forceinline__ float clamp01(float x) { return fminf(fmaxf(x, 0.f), 1.f); }

// ---------------------------------------------------------------------------
// Kernel 1: cur_in[b][n] = sum_k x_t[b][k] * W_in[n][k] + b_in[n]
// ---------------------------------------------------------------------------
__global__ __launch_bounds__(256) void snn_gemm_in(
    const float* __restrict__ x, const float* __restrict__ W_in,
    const float* __restrict__ b_in, float* __restrict__ cur_in)
{
  extern __shared__ float smem[];
  float* Wt2 = smem;                         // [128][WT2_STRIDE]: (W[n][2k2], W[n][2k2+1]) at [k2][2n]
  float* X   = Wt2 + 128 * WT2_STRIDE;       // [32][BUF_STRIDE]
  float* BI  = X + TILE_ROWS * BUF_STRIDE;   // [256]

  const int tid     = threadIdx.x;
  const int lane    = tid & 31;
  const int wave    = tid >> 5;
  const int rowbase = blockIdx.x * TILE_ROWS;

  for (int idx = tid; idx < HID * DIM; idx += 256) {
    int n = idx >> 8, k = idx & 255;
    Wt2[(k >> 1) * WT2_STRIDE + 2 * n + (k & 1)] = W_in[idx];
  }
  BI[tid] = b_in[tid];
  // x has shape (1, B, D, T); we need x[0, b, k, 0] -> flat (b*D + k) * T
  for (int r = 0; r < TILE_ROWS; ++r)
    X[r * BUF_STRIDE + tid] = x[((size_t)(rowbase + r) * DIM + tid) * TSTEPS];
  __syncthreads();

  const int rb    = wave & 1;          // row half of the 32-row tile
  const int cg    = wave >> 1;         // 64-column group
  const int m     = lane & 15;
  const int khalf = (lane >> 4) << 1;  // lanes 16-31 hold K+2, K+3
  const int ln    = lane & 15;
  const int hl    = lane >> 4;

  v8f c[4] = {};
  for (int k0 = 0; k0 < DIM; k0 += 4) {
    const int kk = k0 + khalf;
    const int k2base = (kk >> 1) * WT2_STRIDE;
    v2f a = *(const v2f*)&X[(rb * 16 + m) * BUF_STRIDE + kk];
#pragma unroll
    for (int jj = 0; jj < 4; ++jj) {
      const int n = cg * 64 + jj * 16 + ln;
      v2f b = *(const v2f*)&Wt2[k2base + 2 * n];
      c[jj] = __builtin_amdgcn_wmma_f32_16x16x4_f32(false, a, false, b,
                                                    (short)0, c[jj], false, false);
    }
  }

#pragma unroll
  for (int jj = 0; jj < 4; ++jj) {
#pragma unroll
    for (int j = 0; j < 8; ++j) {
      const int row = rb * 16 + j + hl * 8;
      const int col = cg * 64 + jj * 16 + ln;
      cur_in[(size_t)(rowbase + row) * HID + col] = c[jj][j] + BI[col];
    }
  }
}

// ---------------------------------------------------------------------------
// Kernel 2: fused 64-timestep recurrence for a 32-row batch tile.
// ---------------------------------------------------------------------------
__global__ __launch_bounds__(256) void snn_recurrent(
    const float* __restrict__ cur_in_g,
    const float* __restrict__ W_h,     const float* __restrict__ b_h,
    const float* __restrict__ beta_in, const float* __restrict__ thr_in,
    const float* __restrict__ beta_h,  const float* __restrict__ thr_h,
    const float* __restrict__ W_o,     const float* __restrict__ b_o,
    const float* __restrict__ beta_o,  float* __restrict__ out)
{
  extern __shared__ float smem[];
  float*  Wt2 = smem;                          // [128][WT2_STRIDE]
  float*  BUF = Wt2 + 128 * WT2_STRIDE;        // [32][BUF_STRIDE]  s1 tile, then cur_h tile
  float*  RED = BUF + TILE_ROWS * BUF_STRIDE;  // [8][32] cross-wave reduction
  float4* P   = (float4*)(RED + 256);          // [256] packed (thr_h, clip(beta_h), b_h, W_o)

  const int tid     = threadIdx.x;
  const int lane    = tid & 31;
  const int wave    = tid >> 5;
  const int rowbase = blockIdx.x * TILE_ROWS;

  for (int idx = tid; idx < HID * HID; idx += 256) {
    int n = idx >> 8, k = idx & 255;
    Wt2[(k >> 1) * WT2_STRIDE + 2 * n + (k & 1)] = W_h[idx];
  }
  {
    float4 p;
    p.x = thr_h[tid];
    p.y = clamp01(beta_h[tid]);
    p.z = b_h[tid];
    p.w = W_o[tid];
    P[tid] = p;
  }

  // Layer-1 state: this thread owns hidden unit h = tid for all 32 rows.
  const float thr1 = thr_in[tid];
  const float bb1  = clamp01(beta_in[tid]);
  float m1[TILE_ROWS], ci[TILE_ROWS];
#pragma unroll
  for (int r = 0; r < TILE_ROWS; ++r) {
    ci[r] = cur_in_g[(size_t)(rowbase + r) * HID + tid];
    m1[r] = 0.f;
  }
  // Layer-2 state: this thread owns row (tid & 31), hidden units hg*32 .. hg*32+31.
  const int r2row = tid & 31;
  const int hg    = tid >> 5;
  float m2[32];
#pragma unroll
  for (int j = 0; j < 32; ++j) m2[j] = 0.f;
  float       m3  = 0.f;                 // valid for tid < 32 (one per batch row)
  const float bb3 = clamp01(beta_o[0]);
  const float bo  = b_o[0];

  const int rb    = wave & 1;
  const int cg    = wave >> 1;
  const int m     = lane & 15;
  const int khalf = (lane >> 4) << 1;
  const int ln    = lane & 15;
  const int hl    = lane >> 4;

  __syncthreads();

  for (int t = 0; t < TSTEPS; ++t) {
    // Phase A: m1 update + spike tile s1 -> BUF
#pragma unroll
    for (int r = 0; r < TILE_ROWS; ++r) {
      float mv = m1[r];
      float rr = (mv > thr1) ? thr1 : 0.f;   // reset term r1*thr
      mv = bb1 * mv + ci[r] - rr;
      m1[r] = mv;
      BUF[r * BUF_STRIDE + tid] = (mv > thr1) ? 1.f : 0.f;
    }
    __syncthreads();

    // Phase B: cur_h tile = s1 @ W_h^T via f32 WMMA (W_h resident in LDS)
    v8f c[4] = {};
    for (int k0 = 0; k0 < HID; k0 += 4) {
      const int kk = k0 + khalf;
      const int k2base = (kk >> 1) * WT2_STRIDE;
      v2f a = *(const v2f*)&BUF[(rb * 16 + m) * BUF_STRIDE + kk];
#pragma unroll
      for (int jj = 0; jj < 4; ++jj) {
        const int n = cg * 64 + jj * 16 + ln;
        v2f b = *(const v2f*)&Wt2[k2base + 2 * n];
        c[jj] = __builtin_amdgcn_wmma_f32_16x16x4_f32(false, a, false, b,
                                                      (short)0, c[jj], false, false);
      }
    }
    __syncthreads();

    // Phase C: spill C tiles (cur_h without bias) back into BUF
#pragma unroll
    for (int jj = 0; jj < 4; ++jj) {
#pragma unroll
      for (int j = 0; j < 8; ++j) {
        const int row = rb * 16 + j + hl * 8;
        const int col = cg * 64 + jj * 16 + ln;
        BUF[row * BUF_STRIDE + col] = c[jj][j];
      }
    }
    __syncthreads();

    // Phase D: m2 update, s2, partial dot with W_o (vectorized b128 LDS reads)
    float partial = 0.f;
#pragma unroll
    for (int jv = 0; jv < 8; ++jv) {
      const float4 ch4 = *(const float4*)&BUF[r2row * BUF_STRIDE + hg * 32 + jv * 4];
      const float chv[4] = {ch4.x, ch4.y, ch4.z, ch4.w};
#pragma unroll
      for (int q = 0; q < 4; ++q) {
        const int j = jv * 4 + q;
        const float4 p = P[hg * 32 + j];       // broadcast b128: thr, beta, bias, w_o
        float mv = m2[j];
        float rr = (mv > p.x) ? p.x : 0.f;
        mv = p.y * mv + (chv[q] + p.z) - rr;
        m2[j] = mv;
        partial += (mv > p.x) ? p.w : 0.f;
      }
    }
    RED[hg * 32 + r2row] = partial;
    __syncthreads();

    if (tid < 32) {
      float s = bo;
#pragma unroll
      for (int w = 0; w < 8; ++w) s += RED[w * 32 + tid];
      m3 = bb3 * m3 + s;
    }
    __syncthreads();
  }

  if (tid < 32) atomicAdd(out, m3 * (1.0f / (float)B_TOT));
}

// ---------------------------------------------------------------------------
extern "C" void kernel_launch(void* const* d_in, const int* in_sizes, int n_in,
                              void* d_out, int out_size, void* d_ws, size_t ws_size,
                              hipStream_t stream) {
  (void)in_sizes; (void)n_in; (void)out_size; (void)ws_size;
  const float* x       = (const float*)d_in[0];
  const float* W_in    = (const float*)d_in[1];
  const float* b_in    = (const float*)d_in[2];
  const float* beta_in = (const float*)d_in[3];
  const float* thr_in  = (const float*)d_in[4];
  const float* W_h     = (const float*)d_in[5];
  const float* b_h     = (const float*)d_in[6];
  const float* beta_h  = (const float*)d_in[7];
  const float* thr_h   = (const float*)d_in[8];
  const float* W_o     = (const float*)d_in[9];
  const float* b_o     = (const float*)d_in[10];
  const float* beta_o  = (const float*)d_in[11];
  float* out    = (float*)d_out;
  float* cur_in = (float*)d_ws;   // 8192*256 f32 = 8 MB scratch

  const size_t smem1 = (size_t)(128 * WT2_STRIDE + TILE_ROWS * BUF_STRIDE + 256) * sizeof(float);
  const size_t smem2 = (size_t)(128 * WT2_STRIDE + TILE_ROWS * BUF_STRIDE + 256 + 4 * 256) * sizeof(float);

  hipFuncSetAttribute((const void*)snn_gemm_in,
                      hipFuncAttributeMaxDynamicSharedMemorySize, (int)smem1);
  hipFuncSetAttribute((const void*)snn_recurrent,
                      hipFuncAttributeMaxDynamicSharedMemorySize, (int)smem2);

  hipMemsetAsync(d_out, 0, sizeof(float), stream);

  snn_gemm_in<<<B_TOT / TILE_ROWS, 256, smem1, stream>>>(x, W_in, b_in, cur_in);
  snn_recurrent<<<B_TOT / TILE_ROWS, 256, smem2, stream>>>(
      cur_in, W_h, b_h, beta_in, thr_in, beta_h, thr_h, W_o, b_o, beta_o, out);
}